// GAT_49684181680190
// MI455X (gfx1250) — compile-verified
//
#include <hip/hip_runtime.h>
#include <float.h>

typedef __attribute__((ext_vector_type(2))) float v2f;
typedef __attribute__((ext_vector_type(8))) float v8f;

#define IN_CH 128
#define HIDC  64
#define HEADS 4
#define L1C   256   /* HEADS*HIDC */
#define NEG_SLOPE 0.2f

// ---------------------------------------------------------------------------
// GEMM1: h1[N,256] = x[N,128] @ W1[128,256] using V_WMMA_F32_16X16X4_F32.
// grid = (N/16, 2), block = 256 (8 waves). Wave w handles col tile
// ct = blockIdx.y*8 + w (16 cols). 32 WMMA steps over K=128.
// ---------------------------------------------------------------------------
__global__ void __launch_bounds__(256) gat_gemm1_wmma(
    const float* __restrict__ x, const float* __restrict__ W1,
    float* __restrict__ h1) {
  const int lane = threadIdx.x & 31;
  const int wave = threadIdx.x >> 5;
  const int row0 = blockIdx.x << 4;                 // 16-row tile
  const int n0   = (blockIdx.y * 8 + wave) << 4;    // 16-col tile
  const int m    = lane & 15;
  const int hi   = lane >> 4;                       // lane half selects K pair / row half
  const int nl   = lane & 15;

  v8f acc = {};
  const float* xrow = x + (size_t)(row0 + m) * IN_CH;
  for (int k = 0; k < IN_CH; k += 4) {
    const int kk = k + 2 * hi;
    v2f a, b;
    a.x = xrow[kk];
    a.y = xrow[kk + 1];
    b.x = W1[(size_t)kk * L1C + n0 + nl];
    b.y = W1[(size_t)(kk + 1) * L1C + n0 + nl];
    acc = __builtin_amdgcn_wmma_f32_16x16x4_f32(false, a, false, b,
                                                (short)0, acc, false, false);
  }
#pragma unroll
  for (int r = 0; r < 8; ++r)
    h1[(size_t)(row0 + r + 8 * hi) * L1C + n0 + nl] = acc[r];
}

// ---------------------------------------------------------------------------
// Attention logits layer 1: alS/alD[n,head] = dot(h1[n,head,:], a_src/a_dst)
// ---------------------------------------------------------------------------
__global__ void gat_alpha1(const float* __restrict__ h1,
                           const float* __restrict__ a_src,
                           const float* __restrict__ a_dst,
                           float* __restrict__ alS, float* __restrict__ alD,
                           int N) {
  int tid = blockIdx.x * blockDim.x + threadIdx.x;
  if (tid >= N * HEADS) return;
  int n = tid >> 2, head = tid & 3;
  const float4* hv = (const float4*)(h1 + (size_t)n * L1C + head * HIDC);
  const float4* as = (const float4*)(a_src + head * HIDC);
  const float4* ad = (const float4*)(a_dst + head * HIDC);
  float ss = 0.f, sd = 0.f;
#pragma unroll
  for (int i = 0; i < HIDC / 4; ++i) {
    float4 h4 = hv[i], a4 = as[i], d4 = ad[i];
    ss += h4.x * a4.x + h4.y * a4.y + h4.z * a4.z + h4.w * a4.w;
    sd += h4.x * d4.x + h4.y * d4.y + h4.z * d4.z + h4.w * d4.w;
  }
  alS[tid] = ss;
  alD[tid] = sd;
}

// ---------------------------------------------------------------------------
__global__ void gat_fill(float* __restrict__ p, float v, long n) {
  long i = blockIdx.x * (long)blockDim.x + threadIdx.x;
  if (i < n) p[i] = v;
}

__global__ void gat_init_out(float* __restrict__ o, const float* __restrict__ b2,
                             int N) {
  int i = blockIdx.x * blockDim.x + threadIdx.x;
  if (i < 2 * N) o[i] = b2[i & 1];
}

// ---------------------------------------------------------------------------
// Edge pass 1: segment max of leaky_relu(alS[src]+alD[dst]) per (dst,head).
// Self-loops handled implicitly: edge id >= E -> src=dst=id-E.
// ---------------------------------------------------------------------------
template <int H>
__global__ void gat_edge_max(const int* __restrict__ ei, long E, long NE,
                             const float* __restrict__ alS,
                             const float* __restrict__ alD,
                             float* __restrict__ mArr) {
  long idx = blockIdx.x * (long)blockDim.x + threadIdx.x;
  if (idx >= NE * H) return;
  long e = idx / H;
  int head = (int)(idx - e * H);
  int src, dst;
  if (e < E) { src = ei[e]; dst = ei[E + e]; }
  else       { src = dst = (int)(e - E); }
  float v = alS[(long)src * H + head] + alD[(long)dst * H + head];
  v = v > 0.f ? v : NEG_SLOPE * v;
  atomicMax(&mArr[(long)dst * H + head], v);
}

// ---------------------------------------------------------------------------
// Edge pass 2: p = exp(e - m[dst]); store p per edge, accumulate denom[dst].
// ---------------------------------------------------------------------------
template <int H>
__global__ void gat_edge_exp(const int* __restrict__ ei, long E, long NE,
                             const float* __restrict__ alS,
                             const float* __restrict__ alD,
                             const float* __restrict__ mArr,
                             float* __restrict__ pArr,
                             float* __restrict__ den) {
  long idx = blockIdx.x * (long)blockDim.x + threadIdx.x;
  if (idx >= NE * H) return;
  long e = idx / H;
  int head = (int)(idx - e * H);
  int src, dst;
  if (e < E) { src = ei[e]; dst = ei[E + e]; }
  else       { src = dst = (int)(e - E); }
  float v = alS[(long)src * H + head] + alD[(long)dst * H + head];
  v = v > 0.f ? v : NEG_SLOPE * v;
  float p = __expf(v - mArr[(long)dst * H + head]);
  pArr[idx] = p;
  atomicAdd(&den[(long)dst * H + head], p);
}

// ---------------------------------------------------------------------------
// Edge pass 3 (layer 1): out1[dst,:] += h1[src,:] * alpha  (256 ch per edge).
// One block per edge; per-edge scalars broadcast through LDS.
// ---------------------------------------------------------------------------
__global__ void __launch_bounds__(256) gat_edge_msg1(
    const int* __restrict__ ei, long E,
    const float* __restrict__ h1, const float* __restrict__ pArr,
    const float* __restrict__ den, float* __restrict__ out1) {
  __shared__ int s_src, s_dst;
  __shared__ float s_alpha[HEADS];
  long e = blockIdx.x;
  int t = threadIdx.x;
  if (t == 0) {
    int src, dst;
    if (e < E) { src = ei[e]; dst = ei[E + e]; }
    else       { src = dst = (int)(e - E); }
    s_src = src; s_dst = dst;
#pragma unroll
    for (int hh = 0; hh < HEADS; ++hh)
      s_alpha[hh] = pArr[e * HEADS + hh] /
                    (den[(long)dst * HEADS + hh] + 1e-16f);
  }
  __syncthreads();
  const int src = s_src, dst = s_dst;
  const float a = s_alpha[t >> 6];
  atomicAdd(&out1[(long)dst * L1C + t], h1[(long)src * L1C + t] * a);
}

// ---------------------------------------------------------------------------
// Layer-2 fused: y2[n,0:2] = relu(out1[n,:] + b1) @ W2 ; layer-2 logits.
// ---------------------------------------------------------------------------
__global__ void gat_relu_gemm2(const float* __restrict__ out1,
                               const float* __restrict__ b1,
                               const float* __restrict__ W2,
                               const float* __restrict__ as2,
                               const float* __restrict__ ad2,
                               float* __restrict__ y2,
                               float* __restrict__ alS2,
                               float* __restrict__ alD2, int N) {
  int n = blockIdx.x * blockDim.x + threadIdx.x;
  if (n >= N) return;
  const float* row = out1 + (size_t)n * L1C;
  float acc0 = 0.f, acc1 = 0.f;
  for (int k = 0; k < L1C; k += 4) {
    float4 v4 = *(const float4*)(row + k);
    float4 b4 = *(const float4*)(b1 + k);
    float v;
    v = fmaxf(v4.x + b4.x, 0.f); acc0 += v * W2[(k + 0) * 2]; acc1 += v * W2[(k + 0) * 2 + 1];
    v = fmaxf(v4.y + b4.y, 0.f); acc0 += v * W2[(k + 1) * 2]; acc1 += v * W2[(k + 1) * 2 + 1];
    v = fmaxf(v4.z + b4.z, 0.f); acc0 += v * W2[(k + 2) * 2]; acc1 += v * W2[(k + 2) * 2 + 1];
    v = fmaxf(v4.w + b4.w, 0.f); acc0 += v * W2[(k + 3) * 2]; acc1 += v * W2[(k + 3) * 2 + 1];
  }
  y2[n * 2] = acc0;
  y2[n * 2 + 1] = acc1;
  alS2[n] = acc0 * as2[0] + acc1 * as2[1];
  alD2[n] = acc0 * ad2[0] + acc1 * ad2[1];
}

// ---------------------------------------------------------------------------
// Edge pass 3 (layer 2): d_out[dst,0:2] += y2[src,0:2] * alpha.
// ---------------------------------------------------------------------------
__global__ void gat_edge_msg2(const int* __restrict__ ei, long E, long NE,
                              const float* __restrict__ y2,
                              const float* __restrict__ p2,
                              const float* __restrict__ den2,
                              float* __restrict__ out) {
  long e = blockIdx.x * (long)blockDim.x + threadIdx.x;
  if (e >= NE) return;
  int src, dst;
  if (e < E) { src = ei[e]; dst = ei[E + e]; }
  else       { src = dst = (int)(e - E); }
  float alpha = p2[e] / (den2[dst] + 1e-16f);
  atomicAdd(&out[(long)dst * 2 + 0], y2[(long)src * 2 + 0] * alpha);
  atomicAdd(&out[(long)dst * 2 + 1], y2[(long)src * 2 + 1] * alpha);
}

// ---------------------------------------------------------------------------
extern "C" void kernel_launch(void* const* d_in, const int* in_sizes, int n_in,
                              void* d_out, int out_size, void* d_ws, size_t ws_size,
                              hipStream_t stream) {
  const float* x     = (const float*)d_in[0];
  const int*   ei    = (const int*)d_in[1];
  const float* W1    = (const float*)d_in[2];
  const float* aS1   = (const float*)d_in[3];
  const float* aD1   = (const float*)d_in[4];
  const float* b1    = (const float*)d_in[5];
  const float* W2    = (const float*)d_in[6];
  const float* aS2   = (const float*)d_in[7];
  const float* aD2   = (const float*)d_in[8];
  const float* b2    = (const float*)d_in[9];
  float* out = (float*)d_out;

  const long N  = in_sizes[0] / IN_CH;   // 50000
  const long E  = in_sizes[1] / 2;       // 800000
  const long NE = E + N;                 // with self-loops

  // Workspace carve-out (floats)
  float* ws = (float*)d_ws;
  size_t o = 0;
  float* h1   = ws + o; o += (size_t)N * L1C;
  float* out1 = ws + o; o += (size_t)N * L1C;
  float* p1   = ws + o; o += (size_t)NE * HEADS;
  float* alS1 = ws + o; o += (size_t)N * HEADS;
  float* alD1 = ws + o; o += (size_t)N * HEADS;
  float* m1   = ws + o; o += (size_t)N * HEADS;
  float* den1 = ws + o; o += (size_t)N * HEADS;
  float* y2   = ws + o; o += (size_t)N * 2;
  float* alS2 = ws + o; o += (size_t)N;
  float* alD2 = ws + o; o += (size_t)N;
  float* m2   = ws + o; o += (size_t)N;
  float* den2 = ws + o; o += (size_t)N;
  float* p2   = ws + o; o += (size_t)NE;
  (void)ws_size; (void)n_in; (void)out_size;

  const int B = 256;
  auto blocks = [](long n, int b) { return (unsigned)((n + b - 1) / b); };

  // ---- Layer 1 ----
  gat_gemm1_wmma<<<dim3((unsigned)(N / 16), 2), B, 0, stream>>>(x, W1, h1);
  gat_alpha1<<<blocks(N * HEADS, B), B, 0, stream>>>(h1, aS1, aD1, alS1, alD1, (int)N);
  gat_fill<<<blocks(N * HEADS, B), B, 0, stream>>>(m1, -FLT_MAX, N * HEADS);
  gat_fill<<<blocks(N * HEADS, B), B, 0, stream>>>(den1, 0.f, N * HEADS);
  gat_fill<<<blocks(N * L1C, B), B, 0, stream>>>(out1, 0.f, N * L1C);
  gat_edge_max<HEADS><<<blocks(NE * HEADS, B), B, 0, stream>>>(ei, E, NE, alS1, alD1, m1);
  gat_edge_exp<HEADS><<<blocks(NE * HEADS, B), B, 0, stream>>>(ei, E, NE, alS1, alD1, m1, p1, den1);
  gat_edge_msg1<<<(unsigned)NE, B, 0, stream>>>(ei, E, h1, p1, den1, out1);

  // ---- Layer 2 ----
  gat_relu_gemm2<<<blocks(N, B), B, 0, stream>>>(out1, b1, W2, aS2, aD2, y2, alS2, alD2, (int)N);
  gat_fill<<<blocks(N, B), B, 0, stream>>>(m2, -FLT_MAX, N);
  gat_fill<<<blocks(N, B), B, 0, stream>>>(den2, 0.f, N);
  gat_init_out<<<blocks(2 * N, B), B, 0, stream>>>(out, b2, (int)N);
  gat_edge_max<1><<<blocks(NE, B), B, 0, stream>>>(ei, E, NE, alS2, alD2, m2);
  gat_edge_exp<1><<<blocks(NE, B), B, 0, stream>>>(ei, E, NE, alS2, alD2, m2, p2, den2);
  gat_edge_msg2<<<blocks(NE, B), B, 0, stream>>>(ei, E, NE, y2, p2, den2, out);
}